// Heirachical_Loss_82523501626147
// MI455X (gfx1250) — compile-verified
//
#include <hip/hip_runtime.h>

typedef __attribute__((ext_vector_type(2))) float v2f;
typedef __attribute__((ext_vector_type(8))) float v8f;
typedef int v4i __attribute__((vector_size(4 * sizeof(int))));

#define NCLS            1000
#define ROWS_PER_BLOCK  16
#define QUADS_PER_ROW   250      // 1000 / 4 floats per b128
#define KCHUNKS         250      // 1000 / 4 (WMMA K=4 per op)
#define LDS_STRIDE      1004     // dwords; pad so A-frag ds_load_b64 is bank-conflict-free
#define THREADS         256      // 8 wave32s

// CDNA5 async global->LDS path if this toolchain exposes the builtin.
#if defined(__has_builtin)
#  if __has_builtin(__builtin_amdgcn_global_load_async_to_lds_b128)
#    define HL_ASYNC 1
#  endif
#endif

#define WMMA_F32X4(A, B, C) \
    __builtin_amdgcn_wmma_f32_16x16x4_f32(false, (A), false, (B), (short)0, (C), false, false)

__global__ __launch_bounds__(THREADS)
void hl_main(const float* __restrict__ outputs, const int* __restrict__ target,
             int B, float* __restrict__ sink, int atomicMode)
{
    extern __shared__ float smem[];
    float* rows = smem;                                  // 16 * 1004 dwords
    float* wtot = smem + ROWS_PER_BLOCK * LDS_STRIDE;    // 8 waves * 16 rows

    const int tid  = threadIdx.x;
    const int row0 = blockIdx.x * ROWS_PER_BLOCK;

    // ---- Phase 1: stage 16 rows (64 KB) into LDS, coalesced 16B per lane ----
    for (int i = tid; i < ROWS_PER_BLOCK * QUADS_PER_ROW; i += THREADS) {
        const int r = i / QUADS_PER_ROW;
        const int q = i - r * QUADS_PER_ROW;
        const int grow = row0 + r;
        if (grow < B) {
            const float* gp = outputs + (size_t)grow * NCLS + q * 4;
            float*       lp = rows + r * LDS_STRIDE + q * 4;
#ifdef HL_ASYNC
            __builtin_amdgcn_global_load_async_to_lds_b128(
                (v4i*)(void*)const_cast<float*>(gp), (v4i*)(void*)lp,
                /*offset=*/0, /*cpol=*/0);
#else
            *(float4*)lp = *(const float4*)gp;
#endif
        }
    }
#ifdef HL_ASYNC
#  if __has_builtin(__builtin_amdgcn_s_wait_asynccnt)
    __builtin_amdgcn_s_wait_asynccnt(0);
#  else
    asm volatile("s_wait_asynccnt 0" ::: "memory");
#  endif
#endif
    __syncthreads();

    // ---- Phase 2: row totals on the matrix pipe: D += A(16x4 f32) x ones(4x16) ----
    // A layout (ISA 7.12.2): lanes 0-15 -> M=lane, K=c*4+{0,1}; lanes 16-31 -> M=lane-16, K=c*4+2+{0,1}
    const int lane = tid & 31;
    const int half = lane >> 4;
    const int m    = lane & 15;
    // Wave index as an SGPR: loop bounds become scalar -> uniform loop, EXEC stays all-ones,
    // and the compiler can schedule the unrolled body without EXEC-mask control flow.
    const int uwave = __builtin_amdgcn_readfirstlane(tid >> 5);   // 0..7

    v2f bone; bone[0] = 1.0f; bone[1] = 1.0f;      // B = all-ones: every D column = row sum
    const float* afrag = rows + m * LDS_STRIDE + half * 2;

    v8f acc0 = {}, acc1 = {}, acc2 = {}, acc3 = {};
    int c = uwave;
    // Main: 4 chunks in flight -> 4 batched ds_load_b64, then 4 WMMAs on independent accumulators.
    for (; c + 24 < KCHUNKS; c += 32) {
        v2f a0 = *(const v2f*)(afrag + (c     ) * 4);
        v2f a1 = *(const v2f*)(afrag + (c +  8) * 4);
        v2f a2 = *(const v2f*)(afrag + (c + 16) * 4);
        v2f a3 = *(const v2f*)(afrag + (c + 24) * 4);
        acc0 = WMMA_F32X4(a0, bone, acc0);
        acc1 = WMMA_F32X4(a1, bone, acc1);
        acc2 = WMMA_F32X4(a2, bone, acc2);
        acc3 = WMMA_F32X4(a3, bone, acc3);
    }
    // Tail: at most 3 more chunks for this wave.
    for (; c < KCHUNKS; c += 8) {
        v2f a0 = *(const v2f*)(afrag + c * 4);
        acc0 = WMMA_F32X4(a0, bone, acc0);
    }
    v8f acc = (acc0 + acc1) + (acc2 + acc3);

    // Column N=0 lives in lane 0 (rows 0..7 in acc[0..7]) and lane 16 (rows 8..15).
    if (m == 0) {
#pragma unroll
        for (int r = 0; r < 8; ++r)
            wtot[uwave * 16 + half * 8 + r] = acc[r];
    }
    __syncthreads();

    // ---- Phase 3: per-row tail (group sums + target gather) ----
    float local = 0.0f;
    if (tid < ROWS_PER_BLOCK && (row0 + tid) < B) {
        const int r  = tid;
        const int t  = target[row0 + r];
        const int g1 = t / 10;
        const int g2 = t / 100;
        float total = 0.0f;
#pragma unroll
        for (int w = 0; w < 8; ++w) total += wtot[w * 16 + r];
        const float* rp = rows + r * LDS_STRIDE;
        float s100 = 0.0f;
#pragma unroll 4
        for (int j = 0; j < 100; ++j) s100 += rp[g2 * 100 + j];
        float s10 = 0.0f;
#pragma unroll
        for (int j = 0; j < 10; ++j) s10 += rp[g1 * 10 + j];
        const float ot  = rp[t];
        const float win = 0.5f * total + 0.25f * s100 + 0.125f * s10 + 0.125f * ot;
        local = 1.0f - win;
    }
    if (tid < 32) {                                  // wave32 reduce of 16 partials
        for (int off = 8; off > 0; off >>= 1)
            local += __shfl_down(local, off, 32);
        if (tid == 0) {
            if (atomicMode) atomicAdd(sink, local);
            else            sink[blockIdx.x] = local;
        }
    }
}

__global__ void hl_reduce(const float* __restrict__ partial, int n, float* __restrict__ out)
{
    __shared__ float sbuf[256];
    float s = 0.0f;
    for (int i = threadIdx.x; i < n; i += 256) s += partial[i];
    sbuf[threadIdx.x] = s;
    __syncthreads();
    for (int off = 128; off > 0; off >>= 1) {
        if (threadIdx.x < off) sbuf[threadIdx.x] += sbuf[threadIdx.x + off];
        __syncthreads();
    }
    if (threadIdx.x == 0) out[0] = sbuf[0];
}

__global__ void hl_zero(float* out) { out[0] = 0.0f; }

extern "C" void kernel_launch(void* const* d_in, const int* in_sizes, int n_in,
                              void* d_out, int out_size, void* d_ws, size_t ws_size,
                              hipStream_t stream)
{
    const float* outputs = (const float*)d_in[0];
    const int*   target  = (const int*)d_in[1];
    const int    B       = in_sizes[1];
    float*       out     = (float*)d_out;

    const int nblocks = (B + ROWS_PER_BLOCK - 1) / ROWS_PER_BLOCK;     // 2048 for B=32768
    const size_t smem = (size_t)(ROWS_PER_BLOCK * LDS_STRIDE + 8 * 16) * sizeof(float); // 64768 B

    if (ws_size >= (size_t)nblocks * sizeof(float)) {
        // Deterministic two-pass reduction via workspace.
        float* partial = (float*)d_ws;
        hl_main<<<nblocks, THREADS, smem, stream>>>(outputs, target, B, partial, 0);
        hl_reduce<<<1, 256, 0, stream>>>(partial, nblocks, out);
    } else {
        hl_zero<<<1, 1, 0, stream>>>(out);
        hl_main<<<nblocks, THREADS, smem, stream>>>(outputs, target, B, out, 1);
    }
}